// Attention_61272003445206
// MI455X (gfx1250) — compile-verified
//
#include <hip/hip_runtime.h>

typedef __attribute__((ext_vector_type(2))) float v2f;
typedef __attribute__((ext_vector_type(4))) float v4f;
typedef __attribute__((ext_vector_type(8))) float v8f;

#define B_   4
#define N_   1024
#define DIM_ 1024
#define H_   16
#define DH_  64
#define ROT_ 32
#define MEM_ 16
#define J_   (N_ + MEM_)   /* 1040 */
#define TOPK_ 64
#define MASKV (-3.4028234663852886e+38f)

__device__ __forceinline__ v8f wmma_f32(v2f a, v2f b, v8f c) {
  // D = A(16x4) * B(4x16) + C, fp32 WMMA (CDNA5 V_WMMA_F32_16X16X4_F32)
  return __builtin_amdgcn_wmma_f32_16x16x4_f32(false, a, false, b, (short)0, c,
                                               false, false);
}

// ---------------------------------------------------------------------------
// Tensor Data Mover support (CDNA5 TDM). ROCm 7.2 builtin takes 5 args,
// amdgpu-toolchain (ships amd_gfx1250_TDM.h) takes 6. Guarded fallback.
// ---------------------------------------------------------------------------
#if defined(__has_builtin)
#if __has_builtin(__builtin_amdgcn_tensor_load_to_lds)
#define HAVE_TDM 1
#endif
#endif
#ifndef HAVE_TDM
#define HAVE_TDM 0
#endif

#if __has_include(<hip/amd_detail/amd_gfx1250_TDM.h>)
#define TDM_SIX_ARGS 1
#else
#define TDM_SIX_ARGS 0
#endif

#if HAVE_TDM
typedef unsigned int tdm_u32x4 __attribute__((ext_vector_type(4)));
typedef int tdm_i32x4 __attribute__((ext_vector_type(4)));
typedef int tdm_i32x8 __attribute__((ext_vector_type(8)));

// 2D tile load Global->LDS. dims in 4-byte elements; LDS padding:
// pad_ilog: pad after 2^(pad_ilog+1) DWORDs; pad_amt_m1: (pad DWORDs)-1.
__device__ __forceinline__ void tdm_load_2d(const float* gaddr,
                                            unsigned lds_off,
                                            unsigned tile_d0, unsigned tile_d1,
                                            unsigned tensor_d0,
                                            unsigned tensor_d1,
                                            unsigned long long stride0,
                                            unsigned pad_ilog,
                                            unsigned pad_amt_m1) {
  unsigned long long ga = (unsigned long long)(size_t)(const void*)gaddr;
  tdm_u32x4 g0;
  g0[0] = 1u;                                        // count=1, user mode
  g0[1] = lds_off;                                   // lds_addr (bytes)
  g0[2] = (unsigned)(ga & 0xffffffffull);            // global_addr[31:0]
  g0[3] = (unsigned)((ga >> 32) & 0x01ffffffull) | (2u << 30);  // type=2
  tdm_i32x8 g1;
  g1[0] = (int)((2u << 16) |                         // data_size = 4 bytes
                (1u << 20) |                         // pad_enable
                (pad_ilog << 22) | (pad_amt_m1 << 25));
  g1[1] = (int)((tensor_d0 & 0xffffu) << 16);        // atomic_addr=0 | td0.lo
  g1[2] = (int)(((tensor_d0 >> 16) & 0xffffu) | ((tensor_d1 & 0xffffu) << 16));
  g1[3] = (int)(((tensor_d1 >> 16) & 0xffffu) | ((tile_d0 & 0xffffu) << 16));
  g1[4] = (int)(tile_d1 & 0xffffu);                  // tile_dim1 | tile_dim2=0
  g1[5] = (int)(unsigned)(stride0 & 0xffffffffull);  // dim0 stride lo32
  g1[6] = (int)(unsigned)((stride0 >> 32) & 0xffffull);
  g1[7] = 0;
  tdm_i32x4 z4 = {0, 0, 0, 0};
#if TDM_SIX_ARGS
  tdm_i32x8 z8 = {0, 0, 0, 0, 0, 0, 0, 0};
  __builtin_amdgcn_tensor_load_to_lds(g0, g1, z4, z4, z8, 0);
#else
  __builtin_amdgcn_tensor_load_to_lds(g0, g1, z4, z4, 0);
#endif
}
#endif  // HAVE_TDM

// ---------------------------------------------------------------------------
// Tiled fp32 WMMA GEMM: C[M x Nn] = A[M x K] * B[K x Nn] (+ bias)
// Block = 256 threads = 8 waves; tile 64(M) x 64(N); K in 16-slices.
// TDM path: double-buffered LDS, wave0 issues tensor_load_to_lds for the
// next slice (A 64x16 + B 16x64, LDS padding via descriptor), waits on
// TENSORcnt, barrier, 8 WMMAs per wave per slice.
// Fallback: b128 staging + register double-buffer + global_prefetch.
// ---------------------------------------------------------------------------
__global__ __launch_bounds__(256) void gemm_f32_wmma(
    const float* __restrict__ A, const float* __restrict__ Bm,
    const float* __restrict__ bias, float* __restrict__ C,
    int M, int Nn, int K) {
  const int tid  = threadIdx.x;
  const int wave = tid >> 5;
  const int lane = tid & 31;
  const int m    = lane & 15;
  const int khf  = lane >> 4;          // K-half select per ISA A/B layout
  const int n    = lane & 15;
  const int mrow = (wave & 3) * 16;
  const int ncolw = (wave >> 2) * 32;
  const int mblk = blockIdx.y * 64;
  const int nblk = blockIdx.x * 64;
  v8f c0 = {}, c1 = {};

#if HAVE_TDM
  __shared__ float As[2][64 * 20];   // stride 20: 16 k + 4 pad DWORDs
  __shared__ float Bs[2][16 * 68];   // stride 68: 64 n + 4 pad DWORDs
  const int nsl = K >> 4;
  if (wave == 0) {
    tdm_load_2d(A + (size_t)mblk * K, (unsigned)(size_t)(void*)&As[0][0],
                16u, 64u, (unsigned)K, (unsigned)M, (unsigned)K, 3u, 3u);
    tdm_load_2d(Bm + nblk, (unsigned)(size_t)(void*)&Bs[0][0],
                64u, 16u, (unsigned)Nn, (unsigned)K, (unsigned)Nn, 5u, 3u);
  }
  for (int s = 0; s < nsl; ++s) {
    const float* as = As[s & 1];
    const float* bs = Bs[s & 1];
    __syncthreads();                       // prev compute done: buf^1 free
    if (wave == 0) {
      if (s + 1 < nsl) {
        int k0n = (s + 1) << 4;
        tdm_load_2d(A + (size_t)mblk * K + k0n,
                    (unsigned)(size_t)(void*)&As[(s + 1) & 1][0],
                    16u, 64u, (unsigned)K, (unsigned)M, (unsigned)K, 3u, 3u);
        tdm_load_2d(Bm + (size_t)k0n * Nn + nblk,
                    (unsigned)(size_t)(void*)&Bs[(s + 1) & 1][0],
                    64u, 16u, (unsigned)Nn, (unsigned)K, (unsigned)Nn, 5u, 3u);
        __builtin_amdgcn_s_wait_tensorcnt(2);   // slice s pair has landed
      } else {
        __builtin_amdgcn_s_wait_tensorcnt(0);
      }
    }
    __syncthreads();
#pragma unroll
    for (int ks = 0; ks < 16; ks += 4) {
      v2f a, b0, b1;
      a.x  = as[(mrow + m) * 20 + ks + 2 * khf];
      a.y  = as[(mrow + m) * 20 + ks + 2 * khf + 1];
      b0.x = bs[(ks + 2 * khf) * 68 + ncolw + n];
      b0.y = bs[(ks + 2 * khf + 1) * 68 + ncolw + n];
      b1.x = bs[(ks + 2 * khf) * 68 + ncolw + 16 + n];
      b1.y = bs[(ks + 2 * khf + 1) * 68 + ncolw + 16 + n];
      c0 = wmma_f32(a, b0, c0);
      c1 = wmma_f32(a, b1, c1);
    }
  }
#else
  __shared__ float As1[64 * 20];
  __shared__ float Bs1[16 * 68];
  const int arow = tid >> 2, akq = (tid & 3) * 4;   // 64 x 16 A tile
  const int brow = tid >> 4, bcq = (tid & 15) * 4;  // 16 x 64 B tile
  const float* Aptr = A + (size_t)(mblk + arow) * K + akq;
  const float* Bptr = Bm + (size_t)brow * Nn + nblk + bcq;
  v4f aReg = *(const v4f*)(Aptr);
  v4f bReg = *(const v4f*)(Bptr);
  for (int k0 = 0; k0 < K; k0 += 16) {
    __syncthreads();
    *(v4f*)&As1[arow * 20 + akq] = aReg;
    *(v4f*)&Bs1[brow * 68 + bcq] = bReg;
    __syncthreads();
    if (k0 + 16 < K) {
      aReg = *(const v4f*)(Aptr + k0 + 16);
      bReg = *(const v4f*)(Bptr + (size_t)(k0 + 16) * Nn);
      __builtin_prefetch(Aptr + k0 + 32, 0, 1);
      __builtin_prefetch(Bptr + (size_t)(k0 + 32) * Nn, 0, 1);
    }
#pragma unroll
    for (int ks = 0; ks < 16; ks += 4) {
      v2f a, b0, b1;
      a.x  = As1[(mrow + m) * 20 + ks + 2 * khf];
      a.y  = As1[(mrow + m) * 20 + ks + 2 * khf + 1];
      b0.x = Bs1[(ks + 2 * khf) * 68 + ncolw + n];
      b0.y = Bs1[(ks + 2 * khf + 1) * 68 + ncolw + n];
      b1.x = Bs1[(ks + 2 * khf) * 68 + ncolw + 16 + n];
      b1.y = Bs1[(ks + 2 * khf + 1) * 68 + ncolw + 16 + n];
      c0 = wmma_f32(a, b0, c0);
      c1 = wmma_f32(a, b1, c1);
    }
  }
#endif

  const int col0 = nblk + ncolw + n;
  float bb0 = bias ? bias[col0] : 0.0f;
  float bb1 = bias ? bias[col0 + 16] : 0.0f;
#pragma unroll
  for (int r = 0; r < 8; ++r) {
    int row = mblk + mrow + r + 8 * khf;   // C layout: M = r + 8*laneHalf
    C[(size_t)row * Nn + col0]      = c0[r] + bb0;
    C[(size_t)row * Nn + col0 + 16] = c1[r] + bb1;
  }
}

// ---------------------------------------------------------------------------
// Rotary helpers
// ---------------------------------------------------------------------------
__device__ __forceinline__ float rope_elem(const float* __restrict__ y,
                                           size_t rowbase,
                                           const float* __restrict__ rp,
                                           int i, int d) {
  float val = y[rowbase + d];
  if (d < ROT_) {
    float fr = rp[(size_t)i * ROT_ + d];
    float partner = (d < 16) ? -y[rowbase + d + 16] : y[rowbase + d - 16];
    val = val * cosf(fr) + partner * sinf(fr);
  }
  return val;
}

__global__ __launch_bounds__(256) void rope_q_kernel(
    const float* __restrict__ yq, const float* __restrict__ rp,
    float* __restrict__ qh) {
  int t = blockIdx.x * 256 + threadIdx.x;        // B*H*N*DH = 2^22
  int d = t & 63;
  int i = (t >> 6) & (N_ - 1);
  int h = (t >> 16) & (H_ - 1);
  int b = t >> 20;
  size_t rowbase = ((size_t)(b * N_ + i)) * (H_ * DH_) + h * DH_;
  qh[(((size_t)(b * H_ + h)) * N_ + i) * DH_ + d] = rope_elem(yq, rowbase, rp, i, d);
}

__global__ __launch_bounds__(256) void rope_kv_kernel(
    const float* __restrict__ yk, const float* __restrict__ yv,
    const float* __restrict__ rp, const float* __restrict__ mem_k,
    const float* __restrict__ mem_v, float* __restrict__ kh,
    float* __restrict__ vh) {
  int t = blockIdx.x * 256 + threadIdx.x;        // B*H*J*DH
  const int total = B_ * H_ * J_ * DH_;
  if (t >= total) return;
  int d = t % DH_; int rem = t / DH_;
  int j = rem % J_; rem /= J_;
  int h = rem % H_; int b = rem / H_;
  size_t idx = (((size_t)(b * H_ + h)) * J_ + j) * DH_ + d;
  if (j < MEM_) {
    kh[idx] = mem_k[(h * MEM_ + j) * DH_ + d];
    vh[idx] = mem_v[(h * MEM_ + j) * DH_ + d];
  } else {
    int i = j - MEM_;
    size_t rowbase = ((size_t)(b * N_ + i)) * (H_ * DH_) + h * DH_;
    kh[idx] = rope_elem(yk, rowbase, rp, i, d);
    vh[idx] = rope_elem(yv, rowbase, rp, i, d);
  }
}

// ---------------------------------------------------------------------------
// Scores: pre[b,h,i,j] = scale * sum_d q[i,d]*k[j,d]
// Block = 8 waves, tile 128(i) x 16(j); whole K=64 staged once in LDS.
// ---------------------------------------------------------------------------
__global__ __launch_bounds__(256) void scores_kernel(
    const float* __restrict__ qh, const float* __restrict__ kh,
    float* __restrict__ pre) {
  __shared__ float Qs[128 * 68];
  __shared__ float Ks[16 * 68];
  const int tid = threadIdx.x;
  const int wave = tid >> 5, lane = tid & 31;
  const int m = lane & 15, khf = lane >> 4, n = lane & 15;
  const int bh = blockIdx.z;
  const int iblk = blockIdx.y * 128;
  const int jbase = blockIdx.x * 16;
  const float* q = qh + (size_t)bh * N_ * DH_;
  const float* k = kh + (size_t)bh * J_ * DH_;

#pragma unroll
  for (int it = 0; it < 8; ++it) {               // 128x64 Q tile
    int e = tid + 256 * it;
    int row = e >> 4, cq = (e & 15) * 4;
    *(v4f*)&Qs[row * 68 + cq] = *(const v4f*)&q[(size_t)(iblk + row) * DH_ + cq];
  }
  {                                              // 16x64 K tile
    int row = tid >> 4, cq = (tid & 15) * 4;
    *(v4f*)&Ks[row * 68 + cq] = *(const v4f*)&k[(size_t)(jbase + row) * DH_ + cq];
  }
  __syncthreads();

  v8f c = {};
#pragma unroll
  for (int ks = 0; ks < DH_; ks += 4) {
    v2f a, b;
    a.x = Qs[(wave * 16 + m) * 68 + ks + 2 * khf];
    a.y = Qs[(wave * 16 + m) * 68 + ks + 2 * khf + 1];
    b.x = Ks[n * 68 + ks + 2 * khf];
    b.y = Ks[n * 68 + ks + 2 * khf + 1];
    c = wmma_f32(a, b, c);
  }
  float* dst = pre + (size_t)bh * N_ * J_;
#pragma unroll
  for (int r = 0; r < 8; ++r)
    dst[(size_t)(iblk + wave * 16 + r + 8 * khf) * J_ + jbase + n] = c[r] * 0.125f;
}

// ---------------------------------------------------------------------------
// Wave reductions / order mapping
// ---------------------------------------------------------------------------
__device__ __forceinline__ float wred_max(float v) {
  for (int o = 16; o; o >>= 1) v = fmaxf(v, __shfl_xor(v, o, 32));
  return v;
}
__device__ __forceinline__ float wred_sum(float v) {
  for (int o = 16; o; o >>= 1) v += __shfl_xor(v, o, 32);
  return v;
}
__device__ __forceinline__ int wred_sumi(int v) {
  for (int o = 16; o; o >>= 1) v += __shfl_xor(v, o, 32);
  return v;
}
__device__ __forceinline__ unsigned f2ord(float f) {
  unsigned u = __float_as_uint(f);
  return (u & 0x80000000u) ? ~u : (u | 0x80000000u);
}
__device__ __forceinline__ float ord2f(unsigned o) {
  unsigned u = (o & 0x80000000u) ? (o ^ 0x80000000u) : ~o;
  return __uint_as_float(u);
}

#define NPL 33   /* ceil(1040/32) */

// ---------------------------------------------------------------------------
// Fused head-mix -> causal mask -> top-64 threshold -> softmax -> head-mix.
// One block (16 waves, 512 thr) per (b,i). All 16 pre rows staged in LDS once;
// wave k computes mixed row k in registers, writes attn (post output) back
// into the same LDS buffer (barrier-separated), then applies post_proj mix
// and emits attn2 for the A*V GEMM. Cuts DRAM traffic from 1.64GB -> 0.82GB.
// ---------------------------------------------------------------------------
__global__ __launch_bounds__(512) void mix_topk_softmax_kernel(
    const float* __restrict__ pre, const float* __restrict__ pre_proj,
    const float* __restrict__ post_proj, float* __restrict__ post,
    float* __restrict__ attn2) {
  extern __shared__ float rows[];                // H_ * J_ floats (66.5 KB)
  const int tid = threadIdx.x;
  const int wave = tid >> 5, lane = tid & 31;
  const int bi = blockIdx.x;                     // b*N + i
  const int b = bi >> 10, i = bi & (N_ - 1);
  const int k = wave;                            // this wave's output head

  // stage all 16 heads' score rows (coalesced along j)
  for (int h = 0; h < H_; ++h) {
    const float* src = pre + (((size_t)(b * H_ + h)) * N_ + i) * J_;
    for (int j = tid; j < J_; j += 512) rows[h * J_ + j] = src[j];
  }
  __syncthreads();

  float pc[H_];
#pragma unroll
  for (int h = 0; h < H_; ++h) pc[h] = pre_proj[h * H_ + k];

  const int lim = i + MEM_;                      // causal: j > i+MEM masked
  float vals[NPL];
  unsigned ords[NPL];
#pragma unroll
  for (int t = 0; t < NPL; ++t) {
    int j = lane + 32 * t;
    float v = MASKV;
    if (j < J_ && j <= lim) {
      float acc = 0.f;
#pragma unroll
      for (int h = 0; h < H_; ++h) acc += rows[h * J_ + j] * pc[h];
      v = acc;
    }
    vals[t] = v;
    ords[t] = f2ord(v);
  }
  __syncthreads();                               // LDS reads done; safe to reuse

  // binary search (monotone uint order) for the 64th-largest value
  unsigned ans = 0u;
  for (int bit = 31; bit >= 0; --bit) {
    unsigned cand = ans | (1u << bit);
    int cnt = 0;
#pragma unroll
    for (int t = 0; t < NPL; ++t) cnt += (ords[t] >= cand) ? 1 : 0;
    if (wred_sumi(cnt) >= TOPK_) ans = cand;
  }
  const float vk = ord2f(ans);

  float mx = MASKV;
#pragma unroll
  for (int t = 0; t < NPL; ++t) mx = fmaxf(mx, vals[t]);
  mx = wred_max(mx);
  float s = 0.f;
#pragma unroll
  for (int t = 0; t < NPL; ++t) {
    float v = vals[t];
    float pv = (v < vk) ? 0.f : expf(v - mx);
    vals[t] = pv;
    s += pv;
  }
  s = wred_sum(s);
  const float inv = 1.0f / s;

  float* pdst = post + (((size_t)(b * H_ + k)) * N_ + i) * J_;
#pragma unroll
  for (int t = 0; t < NPL; ++t) {
    int j = lane + 32 * t;
    if (j < J_) {
      float p = vals[t] * inv;
      pdst[j] = p;                               // post_softmax_attn output
      rows[k * J_ + j] = p;                      // stash attn for post-mix
    }
  }
  __syncthreads();

  float qc[H_];
#pragma unroll
  for (int h = 0; h < H_; ++h) qc[h] = post_proj[h * H_ + k];
  float* adst = attn2 + (((size_t)(b * H_ + k)) * N_ + i) * J_;
#pragma unroll
  for (int t = 0; t < NPL; ++t) {
    int j = lane + 32 * t;
    if (j < J_) {
      float acc = 0.f;
#pragma unroll
      for (int h = 0; h < H_; ++h) acc += rows[h * J_ + j] * qc[h];
      adst[j] = acc;
    }
  }
}

// ---------------------------------------------------------------------------
// A*V: out[b,h,i,d] = sum_j attn2[b,h,i,j] * v[b,h,j,d]
// Block = 8 waves, tile 128(i) x 16(d); J staged in 16-slices in LDS with
// register double-buffering. Vs stored transposed so B-fragments are
// contiguous b64 LDS loads.
// ---------------------------------------------------------------------------
__global__ __launch_bounds__(256) void av_kernel(
    const float* __restrict__ attn2, const float* __restrict__ vh,
    float* __restrict__ outflat) {
  __shared__ float Ats[128 * 20];
  __shared__ float Vs[16 * 20];                  // Vs[d_local][k_slot]
  const int tid = threadIdx.x;
  const int wave = tid >> 5, lane = tid & 31;
  const int m = lane & 15, khf = lane >> 4, n = lane & 15;
  const int bh = blockIdx.z, b = bh >> 4, h = bh & 15;
  const int iblk = blockIdx.y * 128;
  const int dbase = blockIdx.x * 16;
  const float* Ar = attn2 + (size_t)bh * N_ * J_;
  const float* Vm = vh + (size_t)bh * J_ * DH_;

  const int arow = tid >> 2, akq = (tid & 3) * 4;
  const float* Ap0 = Ar + (size_t)(iblk + arow) * J_ + akq;
  const float* Ap1 = Ar + (size_t)(iblk + arow + 64) * J_ + akq;
  const int vrow = tid >> 2, vcq = (tid & 3) * 4;   // j-slot, d-group
  const float* Vp = Vm + (size_t)vrow * DH_ + dbase + vcq;

  v4f aR0 = *(const v4f*)(Ap0);
  v4f aR1 = *(const v4f*)(Ap1);
  v4f vR = {};
  if (tid < 64) vR = *(const v4f*)(Vp);

  v8f c = {};
  for (int js = 0; js < J_; js += 16) {
    __syncthreads();
    *(v4f*)&Ats[arow * 20 + akq] = aR0;
    *(v4f*)&Ats[(arow + 64) * 20 + akq] = aR1;
    if (tid < 64) {
#pragma unroll
      for (int qd = 0; qd < 4; ++qd)             // transpose into Vs[d][j]
        Vs[(vcq + qd) * 20 + vrow] = vR[qd];
    }
    __syncthreads();
    if (js + 16 < J_) {
      aR0 = *(const v4f*)(Ap0 + js + 16);
      aR1 = *(const v4f*)(Ap1 + js + 16);
      if (tid < 64) vR = *(const v4f*)(Vp + (size_t)(js + 16) * DH_);
    }
#pragma unroll
    for (int ks = 0; ks < 16; ks += 4) {
      v2f a, bf;
      a.x  = Ats[(wave * 16 + m) * 20 + ks + 2 * khf];
      a.y  = Ats[(wave * 16 + m) * 20 + ks + 2 * khf + 1];
      bf.x = Vs[n * 20 + ks + 2 * khf];
      bf.y = Vs[n * 20 + ks + 2 * khf + 1];
      c = wmma_f32(a, bf, c);
    }
  }
#pragma unroll
  for (int r = 0; r < 8; ++r) {
    int i = iblk + wave * 16 + r + 8 * khf;
    outflat[((size_t)(b * N_ + i)) * (H_ * DH_) + h * DH_ + dbase + n] = c[r];
  }
}

// ---------------------------------------------------------------------------
extern "C" void kernel_launch(void* const* d_in, const int* in_sizes, int n_in,
                              void* d_out, int out_size, void* d_ws,
                              size_t ws_size, hipStream_t stream) {
  const float* x         = (const float*)d_in[0];
  const float* rp        = (const float*)d_in[1];
  const float* Wq        = (const float*)d_in[2];
  const float* Wk        = (const float*)d_in[3];
  const float* Wv        = (const float*)d_in[4];
  const float* Wo        = (const float*)d_in[5];
  const float* bo        = (const float*)d_in[6];
  const float* pre_proj  = (const float*)d_in[7];
  const float* post_proj = (const float*)d_in[8];
  const float* mem_k     = (const float*)d_in[9];
  const float* mem_v     = (const float*)d_in[10];

  float* out  = (float*)d_out;                              // B*N*DIM
  float* pre  = out + (size_t)B_ * N_ * DIM_;               // B*H*N*J
  float* post = pre + (size_t)B_ * H_ * N_ * J_;            // B*H*N*J

  float* ws = (float*)d_ws;
  const size_t szBND = (size_t)B_ * N_ * DIM_;      // 4,194,304
  const size_t szKV  = (size_t)B_ * H_ * J_ * DH_;  // 4,259,840
  float* yq      = ws;                 // reused as outflat after rope
  float* yk      = yq + szBND;
  float* yv      = yk + szBND;
  float* qh      = yv + szBND;
  float* kh      = qh + szBND;
  float* vh      = kh + szKV;
  float* attn2   = vh + szKV;          // 273 MB
  float* outflat = yq;

  dim3 gblk(DIM_ / 64, (B_ * N_) / 64);  // (16, 64) for 4096x1024 GEMMs

  gemm_f32_wmma<<<gblk, 256, 0, stream>>>(x, Wq, nullptr, yq, B_ * N_, DIM_, DIM_);
  gemm_f32_wmma<<<gblk, 256, 0, stream>>>(x, Wk, nullptr, yk, B_ * N_, DIM_, DIM_);
  gemm_f32_wmma<<<gblk, 256, 0, stream>>>(x, Wv, nullptr, yv, B_ * N_, DIM_, DIM_);

  rope_q_kernel<<<(B_ * H_ * N_ * DH_) / 256, 256, 0, stream>>>(yq, rp, qh);
  const int kvtot = B_ * H_ * J_ * DH_;
  rope_kv_kernel<<<(kvtot + 255) / 256, 256, 0, stream>>>(yk, yv, rp, mem_k,
                                                          mem_v, kh, vh);

  scores_kernel<<<dim3(J_ / 16, N_ / 128, B_ * H_), 256, 0, stream>>>(qh, kh, pre);

  mix_topk_softmax_kernel<<<B_ * N_, 512, H_ * J_ * sizeof(float), stream>>>(
      pre, pre_proj, post_proj, post, attn2);

  av_kernel<<<dim3(DH_ / 16, N_ / 128, B_ * H_), 256, 0, stream>>>(attn2, vh,
                                                                   outflat);

  gemm_f32_wmma<<<gblk, 256, 0, stream>>>(outflat, Wo, bo, out, B_ * N_, DIM_, DIM_);
}